// BipartiteNANDGraphLayer_63522566308167
// MI455X (gfx1250) — compile-verified
//
#include <hip/hip_runtime.h>
#include <cstdint>

// BipartiteNANDGraphLayer on MI455X (gfx1250).
//
// Pure memory-bound gather: out[row] = ~(A[i0] OP A[i1]) with OP = | or &.
// 512 MB gathered reads + 256 MB streaming writes -> ~33 us HBM floor.
// Strategy:
//   * one workgroup (256 thr, 8 wave32) per output row
//   * stage both 8 KB gathered rows into LDS with 4x global_load_async_to_lds_b128
//     (CDNA5 async data mover, tracked by ASYNCcnt). INST_OFFSET applies to both
//     the global and the LDS address, so the second 4 KB stage reuses the same
//     address VGPRs with offset:4096.
//   * partial wait (s_wait_asynccnt 2; async loads complete in order) lets the
//     stage-0 compute overlap the stage-1 DMA.
//   * non-temporal b128 stores keep the 192 MB L2 reserved for the 256 MB input
//     table (heavy gather reuse), since the output is write-once.

typedef int v4i __attribute__((ext_vector_type(4)));

#define NUM_WORDS   2048          // int32 words per row
#define THREADS     256
#define STAGE_WORDS (THREADS * 4) // 1024 words (4 KB) per stage, 2 stages/row

__global__ __launch_bounds__(THREADS)
void nand_nor_gather_kernel(const int* __restrict__ bits,            // [N_in][2048]
                            const int* __restrict__ idx,             // [N_out][2] int32
                            const unsigned char* __restrict__ nor,   // [N_out] bool (1B)
                            int* __restrict__ out)                   // [N_out][2048]
{
    __shared__ alignas(16) int lds_a[NUM_WORDS];
    __shared__ alignas(16) int lds_b[NUM_WORDS];

    const unsigned row = blockIdx.x;
    const unsigned t   = threadIdx.x;

    // Uniform per-block scalars (compiler lowers to s_load).
    const unsigned ia = (unsigned)idx[2u * row + 0u];
    const unsigned ib = (unsigned)idx[2u * row + 1u];
    const bool is_nor = nor[row] != 0;

    const int* ga = bits + (size_t)ia * NUM_WORDS + t * 4u;
    const int* gb = bits + (size_t)ib * NUM_WORDS + t * 4u;

    // AS3 byte address = low 32 bits of the generic LDS pointer.
    unsigned la = (unsigned)(uintptr_t)(&lds_a[t * 4u]);
    unsigned lb = (unsigned)(uintptr_t)(&lds_b[t * 4u]);

    // Issue all 4 async DMA loads up front: a0, b0, a1, b1 (in-order completion).
    asm volatile("global_load_async_to_lds_b128 %0, %1, off"
                 :: "v"(la), "v"(ga) : "memory");
    asm volatile("global_load_async_to_lds_b128 %0, %1, off"
                 :: "v"(lb), "v"(gb) : "memory");
    asm volatile("global_load_async_to_lds_b128 %0, %1, off offset:4096"
                 :: "v"(la), "v"(ga) : "memory");
    asm volatile("global_load_async_to_lds_b128 %0, %1, off offset:4096"
                 :: "v"(lb), "v"(gb) : "memory");

    int* orow = out + (size_t)row * NUM_WORDS + t * 4u;

    // ---- stage 0: a0/b0 landed once ASYNCcnt <= 2 ----
    asm volatile("s_wait_asynccnt 2" ::: "memory");
    {
        v4i a = *(const v4i*)(&lds_a[t * 4u]);   // ds_load_b128
        v4i b = *(const v4i*)(&lds_b[t * 4u]);
        v4i r = is_nor ? ~(a | b) : ~(a & b);
        __builtin_nontemporal_store(r, (v4i*)orow);  // b128 store, th:NT
    }

    // ---- stage 1: drain remaining DMA ----
    asm volatile("s_wait_asynccnt 0" ::: "memory");
    {
        v4i a = *(const v4i*)(&lds_a[t * 4u + STAGE_WORDS]);
        v4i b = *(const v4i*)(&lds_b[t * 4u + STAGE_WORDS]);
        v4i r = is_nor ? ~(a | b) : ~(a & b);
        __builtin_nontemporal_store(r, (v4i*)(orow + STAGE_WORDS));
    }
}

extern "C" void kernel_launch(void* const* d_in, const int* in_sizes, int n_in,
                              void* d_out, int out_size, void* d_ws, size_t ws_size,
                              hipStream_t stream) {
    const int*           bits = (const int*)d_in[0];            // int32 bit table
    const int*           idx  = (const int*)d_in[1];            // int32 [N_out][2]
    const unsigned char* nor  = (const unsigned char*)d_in[2];  // jnp bool -> 1 byte
    int*                 out  = (int*)d_out;                    // int32 bit patterns

    const int num_outputs = in_sizes[2];   // nor_mask element count == N_out
    nand_nor_gather_kernel<<<num_outputs, THREADS, 0, stream>>>(bits, idx, nor, out);
}